// DisentangleLayer_81527069213081
// MI455X (gfx1250) — compile-verified
//
#include <hip/hip_runtime.h>
#include <math.h>

typedef __attribute__((ext_vector_type(2))) float v2f;
typedef __attribute__((ext_vector_type(8))) float v8f;

#define N_NODES 100000
#define E_EDGES 3200000
#define IN_DIM  256
#define OUT_DIM 256
#define L_HEADS 4
#define D_HEAD  64
#define BCOLS   80      // 64 emb cols + 8 score cols + 8 zero pad

// ---------------------------------------------------------------------------
// Vectorized zero-fill (harness poisons d_out / d_ws with 0xAA before timing).
// n must be a multiple of 4.
// ---------------------------------------------------------------------------
__global__ void zero_f32v4(float4* __restrict__ p, size_t n4) {
    size_t i = (size_t)blockIdx.x * blockDim.x + threadIdx.x;
    size_t stride = (size_t)gridDim.x * blockDim.x;
    float4 z = make_float4(0.f, 0.f, 0.f, 0.f);
    for (; i < n4; i += stride) p[i] = z;
}

// ---------------------------------------------------------------------------
// Prep: build one combined B matrix Bcat[256 x 80] and bias_cat[80]:
//   cols  0..63 : emb_w                       bias = emb_b
//   cols 64..67 : W_lin @ a_src[l]^T          bias = b_lin.a_src[l] + att_b[l]
//   cols 68..71 : W_lin @ a_dst[l]^T          bias = b_lin.a_dst[l]
//   cols 72..79 : 0                           bias = 0
// (h never materializes: s_src = x@(W_lin@a_src^T) + b_lin.a_src^T, etc.)
// att_w is [4, 512]; a_src = att_w[:, :256], a_dst = att_w[:, 256:]
// ---------------------------------------------------------------------------
__global__ void prep_kernel(const float* __restrict__ W_lin,
                            const float* __restrict__ att_w,
                            const float* __restrict__ att_b,
                            const float* __restrict__ b_lin,
                            const float* __restrict__ emb_w,
                            const float* __restrict__ emb_b,
                            float* __restrict__ Bcat,
                            float* __restrict__ bias_cat) {
    int idx = blockIdx.x * blockDim.x + threadIdx.x;   // 0..20479
    int j = idx / BCOLS, c = idx % BCOLS;
    float v = 0.0f;
    if (c < 64) {
        v = emb_w[j * D_HEAD + c];
    } else if (c < 72) {
        int l = c - 64;                                // 0..7
        const float* arow = (l < 4) ? (att_w + l * 512)
                                    : (att_w + (l - 4) * 512 + 256);
        const float* wrow = W_lin + j * 256;
        #pragma unroll 8
        for (int k = 0; k < 256; ++k) v += wrow[k] * arow[k];
    }
    Bcat[idx] = v;

    if (idx < BCOLS) {
        float s = 0.0f;
        if (idx < 64) {
            s = emb_b[idx];
        } else if (idx < 72) {
            int l = idx - 64;
            const float* arow = (l < 4) ? (att_w + l * 512)
                                        : (att_w + (l - 4) * 512 + 256);
            for (int k = 0; k < 256; ++k) s += b_lin[k] * arow[k];
            if (l < 4) s += att_b[l];
        }
        bias_cat[idx] = s;
    }
}

// ---------------------------------------------------------------------------
// Fused WMMA GEMM: single streaming pass over x (NT hint: x is read once,
// keep L2 for the edge-phase working set).
//   5 waves / block of 16 rows; wave w computes cols [16w, 16w+16) of
//   x @ Bcat + bias_cat. Waves 0-3 -> h_em, wave 4 -> scores[N,16]
//   (cols 0-3 = s_src+att_b, 4-7 = s_dst).
// V_WMMA_F32_16X16X4_F32, fp32 accumulation, K stepped by 4 (64 WMMAs/tile).
// A frag (16x4 f32): lane<16 -> row M=lane, (K0,K1); lane>=16 -> (K2,K3).
// C/D: VGPR r, lane<16 -> M=r; lane>=16 -> M=8+r; col = lane&15.
// All loads are uniform-base + 32-bit lane offset + immediate (no 64-bit
// per-lane address chains in the inner loop).
// ---------------------------------------------------------------------------
__global__ __launch_bounds__(160) void fused_gemm_wmma(
        const float* __restrict__ x,
        const float* __restrict__ Bcat,
        const float* __restrict__ bias_cat,
        float* __restrict__ h_em,
        float* __restrict__ scores) {
    const int wave = threadIdx.x >> 5;   // 0..4 (wave-uniform)
    const int lane = threadIdx.x & 31;
    const int row0 = blockIdx.x * 16;    // N_NODES = 6250 * 16 exactly
    const int mrow = lane & 15;
    const int khi  = lane >> 4;          // 0 or 1
    const int colb = wave * 16;

    // 32-bit element offsets (all buffers < 4G elements)
    const unsigned aoff = (unsigned)(row0 + mrow) * IN_DIM + 2u * (unsigned)khi;
    const unsigned boff = (unsigned)(colb + mrow) + 2u * BCOLS * (unsigned)khi;

    v8f c = {};
    #pragma unroll 4
    for (int k0 = 0; k0 < IN_DIM; k0 += 4) {
        // streaming (nontemporal) 64-bit A load: x[row, k0+2*khi .. +1]
        v2f a = __builtin_nontemporal_load((const v2f*)(x + aoff + (unsigned)k0));
        // B loads: rows k0+2*khi, k0+2*khi+1 of Bcat -> constant imm offsets
        v2f b;
        b.x = Bcat[boff + (unsigned)(k0 * BCOLS)];
        b.y = Bcat[boff + (unsigned)(k0 * BCOLS) + BCOLS];
        c = __builtin_amdgcn_wmma_f32_16x16x4_f32(
                /*neg_a=*/false, a, /*neg_b=*/false, b,
                /*c_mod=*/(short)0, c, /*reuse_a=*/false, /*reuse_b=*/false);
    }

    const float bias = bias_cat[colb + mrow];
    const int m0 = khi * 8;
    if (wave < 4) {
        float* outp = h_em + (unsigned)row0 * D_HEAD + colb + mrow;
        #pragma unroll
        for (int r = 0; r < 8; ++r)
            outp[(unsigned)(m0 + r) * D_HEAD] = c[r] + bias;
    } else {
        float* outp = scores + (unsigned)row0 * 16 + mrow;
        #pragma unroll
        for (int r = 0; r < 8; ++r)
            outp[(unsigned)(m0 + r) * 16] = c[r] + bias;
    }
}

__device__ __forceinline__ float edge_w(float zsrc, float zdst) {
    // exp(sigmoid(z)); the reference's global-max shift cancels in attn
    // (exp(e-m)/sum(exp(e-m)) == exp(e)/sum(exp(e)); sigmoid in (0,1) so
    // exp(e) in (1,e) -- no overflow).
    float z = zsrc + zdst;
    float sg = 1.0f / (1.0f + __expf(-z));
    return __expf(sg);
}

// ---------------------------------------------------------------------------
// Edge pass 1: softmax denominators per (src node, head). denom (1.6 MB) and
// scores (6.4 MB) are L2-resident.
// ---------------------------------------------------------------------------
__global__ __launch_bounds__(256) void edge_denom(
        const int* __restrict__ src, const int* __restrict__ dst,
        const float* __restrict__ scores, float* __restrict__ denom) {
    int e = blockIdx.x * blockDim.x + threadIdx.x;
    if (e >= E_EDGES) return;
    unsigned s = (unsigned)src[e], d = (unsigned)dst[e];
    const float4 ss = *(const float4*)(scores + s * 16u);
    const float4 sd = *(const float4*)(scores + d * 16u + 4u);
    float* dn = denom + s * 4u;
    atomicAdd(dn + 0, edge_w(ss.x, sd.x));
    atomicAdd(dn + 1, edge_w(ss.y, sd.y));
    atomicAdd(dn + 2, edge_w(ss.z, sd.z));
    atomicAdd(dn + 3, edge_w(ss.w, sd.w));
}

// ---------------------------------------------------------------------------
// Edge pass 2: wave-per-edge scatter. Lane i handles dims {i, i+32} of the
// 64-wide h_em[dst] row; 4 heads -> 8 fp32 atomics/lane into L2-resident out.
// ---------------------------------------------------------------------------
__global__ __launch_bounds__(256) void edge_scatter(
        const int* __restrict__ src, const int* __restrict__ dst,
        const float* __restrict__ scores, const float* __restrict__ denom,
        const float* __restrict__ h_em, float* __restrict__ out) {
    int gid  = blockIdx.x * blockDim.x + threadIdx.x;
    int e    = gid >> 5;
    int lane = gid & 31;
    if (e >= E_EDGES) return;
    unsigned s = (unsigned)src[e], d = (unsigned)dst[e];

    const float4 ss = *(const float4*)(scores + s * 16u);
    const float4 sd = *(const float4*)(scores + d * 16u + 4u);
    const float4 dn = *(const float4*)(denom + s * 4u);
    float a0 = edge_w(ss.x, sd.x) / dn.x;
    float a1 = edge_w(ss.y, sd.y) / dn.y;
    float a2 = edge_w(ss.z, sd.z) / dn.z;
    float a3 = edge_w(ss.w, sd.w) / dn.w;

    const float* hv = h_em + d * (unsigned)D_HEAD;
    float h0 = hv[lane];
    float h1 = hv[lane + 32];

    float* ob = out + s * (unsigned)OUT_DIM;
    atomicAdd(ob +   0 + lane, a0 * h0);  atomicAdd(ob +  32 + lane, a0 * h1);
    atomicAdd(ob +  64 + lane, a1 * h0);  atomicAdd(ob +  96 + lane, a1 * h1);
    atomicAdd(ob + 128 + lane, a2 * h0);  atomicAdd(ob + 160 + lane, a2 * h1);
    atomicAdd(ob + 192 + lane, a3 * h0);  atomicAdd(ob + 224 + lane, a3 * h1);
}

// ---------------------------------------------------------------------------
extern "C" void kernel_launch(void* const* d_in, const int* in_sizes, int n_in,
                              void* d_out, int out_size, void* d_ws, size_t ws_size,
                              hipStream_t stream) {
    const float* x     = (const float*)d_in[0];
    const int*   src   = (const int*)  d_in[1];
    const int*   dst   = (const int*)  d_in[2];
    const float* W_lin = (const float*)d_in[3];
    const float* b_lin = (const float*)d_in[4];
    const float* att_w = (const float*)d_in[5];
    const float* att_b = (const float*)d_in[6];
    const float* emb_w = (const float*)d_in[7];
    const float* emb_b = (const float*)d_in[8];
    float* out = (float*)d_out;

    // Workspace layout (floats)
    float* Bcat     = (float*)d_ws;                        // 256*80 = 20480
    float* bias_cat = Bcat + 256 * BCOLS;                  // 80
    float* scores   = bias_cat + BCOLS;                    // N*16
    float* h_em     = scores + (size_t)N_NODES * 16;       // N*64
    float* denom    = h_em + (size_t)N_NODES * D_HEAD;     // N*4

    // Zero accumulators (out + denom) every call.
    zero_f32v4<<<2048, 256, 0, stream>>>((float4*)out,
                                         (size_t)N_NODES * OUT_DIM / 4);
    zero_f32v4<<<128, 256, 0, stream>>>((float4*)denom,
                                        (size_t)N_NODES * L_HEADS / 4);

    // Build combined B matrix (emb_w | folded score projections).
    prep_kernel<<<BCOLS, 256, 0, stream>>>(W_lin, att_w, att_b, b_lin,
                                           emb_w, emb_b, Bcat, bias_cat);

    // Single fused streaming pass over x: h_em + scores via fp32 WMMA.
    fused_gemm_wmma<<<N_NODES / 16, 160, 0, stream>>>(
        x, Bcat, bias_cat, h_em, scores);

    // Edge phase.
    edge_denom<<<E_EDGES / 256, 256, 0, stream>>>(src, dst, scores, denom);
    edge_scatter<<<(E_EDGES * 32) / 256, 256, 0, stream>>>(
        src, dst, scores, denom, h_em, out);
}